// RelationMessagePassing_10385230921929
// MI455X (gfx1250) — compile-verified
//
#include <hip/hip_runtime.h>
#include <hip/hip_bf16.h>

typedef __attribute__((ext_vector_type(2))) float v2f;
typedef __attribute__((ext_vector_type(8))) float v8f;

__device__ __forceinline__ v8f wmma4(v2f a, v2f b, v8f c) {
  // D = A(16x4 f32) * B(4x16 f32) + C(16x16 f32)  -> v_wmma_f32_16x16x4_f32
  return __builtin_amdgcn_wmma_f32_16x16x4_f32(false, a, false, b, (short)0, c,
                                               false, false);
}

// monotone float<->uint mapping so atomicMax(uint) implements float max
__device__ __forceinline__ unsigned f2key(float f) {
  unsigned u = __float_as_uint(f);
  return (u & 0x80000000u) ? ~u : (u | 0x80000000u);
}
__device__ __forceinline__ float key2f(unsigned k) {
  unsigned u = (k & 0x80000000u) ? (k ^ 0x80000000u) : ~k;
  return __uint_as_float(u);
}

__global__ void init_kernel(float* __restrict__ out, int total,
                            unsigned* __restrict__ max_key) {
  int i = blockIdx.x * blockDim.x + threadIdx.x;
  if (i == 0) *max_key = 0u;
  if (i < total) out[i] = 1e-16f;
}

// One relation MLP pass over a 64-tuple tile.
// mode 0: compute o = W2*relu(W1*x+b1)+b2, reduce global max into max_key.
// mode 1: recompute o, scatter-add exp(8*(o - maxoff)) into exps_sum[idx].
template <int D>
__global__ __launch_bounds__(256) void rel_mlp_kernel(
    const float* __restrict__ node_states, const int* __restrict__ idx,
    const float* __restrict__ w1, const float* __restrict__ b1,
    const float* __restrict__ w2, const float* __restrict__ b2, int T, int mode,
    unsigned* __restrict__ max_key, float* __restrict__ exps_sum) {
  constexpr int ARITY = D / 64;
  constexpr int STRIDE = D + 4;  // pad: conflict-free A-frag LDS reads
  constexpr int TPW = D / 32;    // 16x16 C tiles per wave (half the columns)
  __shared__ float sm[64 * STRIDE];
  __shared__ unsigned smax;

  const int tid = threadIdx.x;
  const int lane = tid & 31;
  const int wave = tid >> 5;
  const int t0 = blockIdx.x * 64;

  if (tid == 0) smax = 0u;
  const float maxoff = (mode == 1) ? key2f(*max_key) : 0.0f;

  // Gather X tile [64, D] into LDS (coalesced 64-float runs per node row)
  for (int i = tid; i < 64 * D; i += 256) {
    int row = i / D, col = i % D;
    float v = 0.0f;
    int t = t0 + row;
    if (t < T) {
      int node = idx[t * ARITY + (col >> 6)];
      v = node_states[node * 64 + (col & 63)];
    }
    sm[row * STRIDE + col] = v;
  }
  __syncthreads();

  const int m0 = (wave & 3) * 16;      // 16-row stripe
  const int n0base = (wave >> 2) * (D / 2);
  const int aRow = lane & 15;
  const int kh = (lane >> 4) * 2;      // lanes 16-31 hold K+2,K+3

  // ---- GEMM1: H = relu(X * W1^T + b1)
  v8f acc[TPW];
#pragma unroll
  for (int j = 0; j < TPW; ++j)
#pragma unroll
    for (int e = 0; e < 8; ++e) acc[j][e] = 0.0f;

  for (int k0 = 0; k0 < D; k0 += 4) {
    v2f a;
    a.x = sm[(m0 + aRow) * STRIDE + k0 + kh];
    a.y = sm[(m0 + aRow) * STRIDE + k0 + kh + 1];
#pragma unroll
    for (int j = 0; j < TPW; ++j) {
      int n = n0base + j * 16 + (lane & 15);
      v2f b;  // B[k][n] = w1[n][k] (weights are L2/WGP$-resident)
      b.x = w1[n * D + k0 + kh];
      b.y = w1[n * D + k0 + kh + 1];
      acc[j] = wmma4(a, b, acc[j]);
    }
  }
  __syncthreads();  // all waves done reading X; reuse LDS for H
#pragma unroll
  for (int j = 0; j < TPW; ++j) {
    int n = n0base + j * 16 + (lane & 15);
    float bias = b1[n];
#pragma unroll
    for (int e = 0; e < 8; ++e) {
      int m = m0 + e + ((lane >> 4) << 3);  // C/D layout: VGPR e -> row e / e+8
      float h = acc[j][e] + bias;
      sm[m * STRIDE + n] = h > 0.0f ? h : 0.0f;
    }
  }
  __syncthreads();

  // ---- GEMM2: O = H * W2^T + b2
#pragma unroll
  for (int j = 0; j < TPW; ++j)
#pragma unroll
    for (int e = 0; e < 8; ++e) acc[j][e] = 0.0f;

  for (int k0 = 0; k0 < D; k0 += 4) {
    v2f a;
    a.x = sm[(m0 + aRow) * STRIDE + k0 + kh];
    a.y = sm[(m0 + aRow) * STRIDE + k0 + kh + 1];
#pragma unroll
    for (int j = 0; j < TPW; ++j) {
      int n = n0base + j * 16 + (lane & 15);
      v2f b;
      b.x = w2[n * D + k0 + kh];
      b.y = w2[n * D + k0 + kh + 1];
      acc[j] = wmma4(a, b, acc[j]);
    }
  }

  if (mode == 0) {
    float lmax = -3.4e38f;
#pragma unroll
    for (int j = 0; j < TPW; ++j) {
      int n = n0base + j * 16 + (lane & 15);
      float bias = b2[n];
#pragma unroll
      for (int e = 0; e < 8; ++e) {
        int m = m0 + e + ((lane >> 4) << 3);
        if (t0 + m < T) lmax = fmaxf(lmax, acc[j][e] + bias);
      }
    }
    atomicMax(&smax, f2key(lmax));
    __syncthreads();
    if (tid == 0) atomicMax(max_key, smax);
  } else {
#pragma unroll
    for (int j = 0; j < TPW; ++j) {
      int n = n0base + j * 16 + (lane & 15);
      int g = n >> 6, col = n & 63;
      float bias = b2[n];
#pragma unroll
      for (int e = 0; e < 8; ++e) {
        int m = m0 + e + ((lane >> 4) << 3);
        int t = t0 + m;
        if (t < T) {
          int node = idx[t * ARITY + g];
          float o = acc[j][e] + bias;
          atomicAdd(&exps_sum[node * 64 + col], expf(8.0f * (o - maxoff)));
        }
      }
    }
  }
}

// Final update MLP: cat = [0.125*log(exps_sum)+maxoff, node_states] [64,128]
// h = relu(cat*uw1^T+ub1); out = h*uw2^T+ub2.  `out` holds exps_sum on entry
// and final output on exit (row-local read-before-write within each block).
__global__ __launch_bounds__(256) void update_kernel(
    const float* __restrict__ node_states, const unsigned* __restrict__ max_key,
    const float* __restrict__ uw1, const float* __restrict__ ub1,
    const float* __restrict__ uw2, const float* __restrict__ ub2, int Nn,
    float* __restrict__ out) {
  constexpr int STRIDE = 132;
  __shared__ float sm[64 * STRIDE];
  const int tid = threadIdx.x, lane = tid & 31, wave = tid >> 5;
  const int r0 = blockIdx.x * 64;
  const float maxoff = key2f(*max_key);

  for (int i = tid; i < 64 * 128; i += 256) {
    int row = i / 128, col = i % 128;
    int r = r0 + row;
    float v = 0.0f;
    if (r < Nn) {
      if (col < 64) v = 0.125f * logf(out[r * 64 + col]) + maxoff;
      else          v = node_states[r * 64 + (col - 64)];
    }
    sm[row * STRIDE + col] = v;
  }
  __syncthreads();

  const int m0 = (wave & 3) * 16;
  const int cg = wave >> 2;
  const int aRow = lane & 15;
  const int kh = (lane >> 4) * 2;

  // GEMM1: [64,128] x uw1^T[128,128]
  v8f acc[4];
#pragma unroll
  for (int j = 0; j < 4; ++j)
#pragma unroll
    for (int e = 0; e < 8; ++e) acc[j][e] = 0.0f;
  for (int k0 = 0; k0 < 128; k0 += 4) {
    v2f a;
    a.x = sm[(m0 + aRow) * STRIDE + k0 + kh];
    a.y = sm[(m0 + aRow) * STRIDE + k0 + kh + 1];
#pragma unroll
    for (int j = 0; j < 4; ++j) {
      int n = cg * 64 + j * 16 + (lane & 15);
      v2f b;
      b.x = uw1[n * 128 + k0 + kh];
      b.y = uw1[n * 128 + k0 + kh + 1];
      acc[j] = wmma4(a, b, acc[j]);
    }
  }
  __syncthreads();
#pragma unroll
  for (int j = 0; j < 4; ++j) {
    int n = cg * 64 + j * 16 + (lane & 15);
    float bias = ub1[n];
#pragma unroll
    for (int e = 0; e < 8; ++e) {
      int m = m0 + e + ((lane >> 4) << 3);
      float h = acc[j][e] + bias;
      sm[m * STRIDE + n] = h > 0.0f ? h : 0.0f;
    }
  }
  __syncthreads();

  // GEMM2: [64,128] x uw2^T[128,64] -> [64,64]
  v8f acc2[2];
#pragma unroll
  for (int j = 0; j < 2; ++j)
#pragma unroll
    for (int e = 0; e < 8; ++e) acc2[j][e] = 0.0f;
  for (int k0 = 0; k0 < 128; k0 += 4) {
    v2f a;
    a.x = sm[(m0 + aRow) * STRIDE + k0 + kh];
    a.y = sm[(m0 + aRow) * STRIDE + k0 + kh + 1];
#pragma unroll
    for (int j = 0; j < 2; ++j) {
      int n = cg * 32 + j * 16 + (lane & 15);
      v2f b;
      b.x = uw2[n * 128 + k0 + kh];
      b.y = uw2[n * 128 + k0 + kh + 1];
      acc2[j] = wmma4(a, b, acc2[j]);
    }
  }
#pragma unroll
  for (int j = 0; j < 2; ++j) {
    int n = cg * 32 + j * 16 + (lane & 15);
    float bias = ub2[n];
#pragma unroll
    for (int e = 0; e < 8; ++e) {
      int m = m0 + e + ((lane >> 4) << 3);
      int r = r0 + m;
      if (r < Nn) out[r * 64 + n] = acc2[j][e] + bias;
    }
  }
}

extern "C" void kernel_launch(void* const* d_in, const int* in_sizes, int n_in,
                              void* d_out, int out_size, void* d_ws,
                              size_t ws_size, hipStream_t stream) {
  const float* node_states = (const float*)d_in[0];
  const int* idxs[3] = {(const int*)d_in[1], (const int*)d_in[2],
                        (const int*)d_in[3]};
  const float* rw1[3] = {(const float*)d_in[4], (const float*)d_in[8],
                         (const float*)d_in[12]};
  const float* rb1[3] = {(const float*)d_in[5], (const float*)d_in[9],
                         (const float*)d_in[13]};
  const float* rw2[3] = {(const float*)d_in[6], (const float*)d_in[10],
                         (const float*)d_in[14]};
  const float* rb2[3] = {(const float*)d_in[7], (const float*)d_in[11],
                         (const float*)d_in[15]};
  const float* uw1 = (const float*)d_in[16];
  const float* ub1 = (const float*)d_in[17];
  const float* uw2 = (const float*)d_in[18];
  const float* ub2 = (const float*)d_in[19];

  const int Nn = in_sizes[0] / 64;
  const int T[3] = {in_sizes[1] / 2, in_sizes[2] / 2, in_sizes[3] / 3};

  float* out = (float*)d_out;          // doubles as exps_sum accumulator
  unsigned* max_key = (unsigned*)d_ws; // ordered-uint global max

  const int total = Nn * 64;
  init_kernel<<<(total + 255) / 256, 256, 0, stream>>>(out, total, max_key);

  // pass 1: global max of all messages
  rel_mlp_kernel<128><<<(T[0] + 63) / 64, 256, 0, stream>>>(
      node_states, idxs[0], rw1[0], rb1[0], rw2[0], rb2[0], T[0], 0, max_key, out);
  rel_mlp_kernel<128><<<(T[1] + 63) / 64, 256, 0, stream>>>(
      node_states, idxs[1], rw1[1], rb1[1], rw2[1], rb2[1], T[1], 0, max_key, out);
  rel_mlp_kernel<192><<<(T[2] + 63) / 64, 256, 0, stream>>>(
      node_states, idxs[2], rw1[2], rb1[2], rw2[2], rb2[2], T[2], 0, max_key, out);

  // pass 2: recompute + exp scatter-add (GEMMs are cheap; avoids 460MB staging)
  rel_mlp_kernel<128><<<(T[0] + 63) / 64, 256, 0, stream>>>(
      node_states, idxs[0], rw1[0], rb1[0], rw2[0], rb2[0], T[0], 1, max_key, out);
  rel_mlp_kernel<128><<<(T[1] + 63) / 64, 256, 0, stream>>>(
      node_states, idxs[1], rw1[1], rb1[1], rw2[1], rb2[1], T[1], 1, max_key, out);
  rel_mlp_kernel<192><<<(T[2] + 63) / 64, 256, 0, stream>>>(
      node_states, idxs[2], rw1[2], rb1[2], rw2[2], rb2[2], T[2], 1, max_key, out);

  // final update MLP (in-place on d_out)
  update_kernel<<<(Nn + 63) / 64, 256, 0, stream>>>(node_states, max_key, uw1,
                                                    ub1, uw2, ub2, Nn, out);
}